// Glm4vMoeTextTopkRouter_38585986187310
// MI455X (gfx1250) — compile-verified
//
#include <hip/hip_runtime.h>
#include <hip/hip_bf16.h>
#include <cstddef>
#include <cstdint>

// MoE router for GLM4v: logits = h @ W^T (T=16384, H=4096, E=128), sigmoid,
// +bias, per-token top-8 (group logic is identity for N_GROUP=1), weights =
// raw sigmoid scores at the picked experts normalized to sum 1.
//
// Memory-bound: h = 256 MB read once -> ~11 us floor at 23.3 TB/s.
// Matmul on V_WMMA_F32_16X16X4_F32 (exact fp32, matches reference einsum).
// h tiles are staged into LDS by the Tensor Data Mover (double-buffered,
// one DMA per K-chunk issued by wave 0, TENSORcnt-synchronized), with TDM
// hardware padding producing the bank-conflict-free 132-DWORD row stride.

typedef __attribute__((ext_vector_type(2))) float v2f;
typedef __attribute__((ext_vector_type(8))) float v8f;
typedef __attribute__((ext_vector_type(4))) unsigned int v4u;
typedef __attribute__((ext_vector_type(8))) int v8i;
typedef __attribute__((ext_vector_type(4))) int v4i;

#define HIDDEN    4096
#define NEXP      128
#define TOPK      8
#define BM        32        // tokens per block (2 WMMA M-tiles)
#define KCHUNK    128       // K-chunk staged in LDS
#define NCHUNK    (HIDDEN / KCHUNK)
#define AS_STRIDE 132       // 128 + 4 pad -> A-fragment ds_load_b64 hits all 64 banks
#define SC_STRIDE 130       // 128 + 2 pad -> conflict-free score writes/reads

#if defined(__has_builtin)
#if __has_builtin(__builtin_amdgcn_tensor_load_to_lds) && __has_builtin(__builtin_amdgcn_s_wait_tensorcnt)
#define USE_TDM 1
#endif
#endif
#ifndef USE_TDM
#define USE_TDM 0
#endif

#if USE_TDM
// Issue one TDM descriptor: copy a BM x KCHUNK fp32 tile (row stride HIDDEN)
// from global memory to LDS at lds_byte_off, padding 4 DWORDs after every
// 128 DWORDs (=> LDS row stride AS_STRIDE=132 DWORDs).
// This toolchain's builtin takes 6 args (g0, g1, g2, g3, g4, cpol); trailing
// groups are zero for a plain 2-D tile (tile_dim2=0 => dims 3/4 unused).
__device__ __forceinline__ void tdm_load_tile(const float* gptr, uint32_t lds_byte_off) {
    const unsigned long long ga = (unsigned long long)(uintptr_t)gptr;
    v4u g0;
    g0[0] = 1u;                                     // count=1, user descriptor
    g0[1] = lds_byte_off;                           // lds_addr [63:32]
    g0[2] = (unsigned int)(ga & 0xffffffffu);       // global_addr [95:64]
    g0[3] = (unsigned int)((ga >> 32) & 0x01ffffffu)// global_addr [120:96]
          | (2u << 30);                             // type=2 ("image") [127:126]
    v8i g1;
    g1[0] = (int)((2u << 16)    // data_size = 4 bytes
                | (1u << 20)    // pad_enable
                | (6u << 22)    // pad_interval: 2^(6+1)=128 DWORDs
                | (3u << 25));  // pad_amount:   3+1 = 4 DWORDs
    g1[1] = (int)(((unsigned)KCHUNK) << 16);        // tensor_dim0[15:0] @ bits 63:48
    g1[2] = (int)(((unsigned)BM) << 16);            // dim0 hi=0 | tensor_dim1[15:0]
    g1[3] = (int)(((unsigned)KCHUNK) << 16);        // dim1 hi=0 | tile_dim0=128
    g1[4] = BM;                                     // tile_dim1=32 | tile_dim2=0
    g1[5] = HIDDEN;                                 // tensor_dim0_stride lo32
    g1[6] = 0;                                      // stride hi | dim1_stride lo
    g1[7] = 0;
    const v4i z4 = {0, 0, 0, 0};                    // 2-D tensor: groups 2/3 unused
    const v8i z8 = {0, 0, 0, 0, 0, 0, 0, 0};
    __builtin_amdgcn_tensor_load_to_lds(g0, g1, z4, z4, z8, 0);
}
#endif

__global__ __launch_bounds__(256, 1)
void glm4v_moe_router_kernel(const float* __restrict__ hs,
                             const float* __restrict__ wgt,
                             const float* __restrict__ bias,
                             int*   __restrict__ out_idx,
                             float* __restrict__ out_w)
{
    __shared__ __align__(16) float As[2][BM * AS_STRIDE]; // double-buffered h tile
    __shared__ float Sc[BM * SC_STRIDE];                  // sigmoid scores [tok][exp]
    __shared__ float Bs[NEXP];                            // bias
    __shared__ int   PkI[BM][TOPK];
    __shared__ float PkW[BM][TOPK];

    const int tid  = threadIdx.x;
    const int wave = tid >> 5;               // 8 waves, one 16-expert N-tile each
    const int lane = tid & 31;
    const int lm   = lane & 15;
    const int lh   = lane >> 4;              // K half-select for A/B fragments
    const int m0   = blockIdx.x * BM;
    const int n0   = wave * 16;

    if (tid < NEXP) Bs[tid] = bias[tid];

    v8f acc0 = {};                           // tokens m0+0..15  x experts n0..n0+15
    v8f acc1 = {};                           // tokens m0+16..31 x experts n0..n0+15

    // B fragment base: lane lm holds expert row (n0+lm), K offset 2*lh (+kk).
    const float* wrow = wgt + (size_t)(n0 + lm) * HIDDEN + 2 * lh;
    const float* hbase = hs + (size_t)m0 * HIDDEN;

#if USE_TDM
    if (wave == 0)
        tdm_load_tile(hbase, (uint32_t)(uintptr_t)&As[0][0]);

    for (int c = 0; c < NCHUNK; ++c) {
        if (wave == 0) __builtin_amdgcn_s_wait_tensorcnt(0); // buf[c&1] ready
        __syncthreads();                                     // release all waves
        if (wave == 0 && (c + 1) < NCHUNK)                   // prefetch next chunk
            tdm_load_tile(hbase + (size_t)(c + 1) * KCHUNK,
                          (uint32_t)(uintptr_t)&As[(c + 1) & 1][0]);
        const float* Ab = As[c & 1];
        const int kc = c * KCHUNK;
#pragma unroll 8
        for (int kk = 0; kk < KCHUNK; kk += 4) {
            v2f b  = *(const v2f*)(wrow + kc + kk);
            v2f a0 = *(const v2f*)(Ab + (     lm) * AS_STRIDE + kk + 2 * lh);
            v2f a1 = *(const v2f*)(Ab + (16 + lm) * AS_STRIDE + kk + 2 * lh);
            acc0 = __builtin_amdgcn_wmma_f32_16x16x4_f32(false, a0, false, b,
                                                         (short)0, acc0, false, false);
            acc1 = __builtin_amdgcn_wmma_f32_16x16x4_f32(false, a1, false, b,
                                                         (short)0, acc1, false, false);
        }
    }
#else
    for (int c = 0; c < NCHUNK; ++c) {
        const int kc = c * KCHUNK;
        __syncthreads();
#pragma unroll
        for (int i = 0; i < (BM * KCHUNK / 4) / 256; ++i) {
            int seg = tid + i * 256;
            int row = seg >> 5;
            int c4  = (seg & 31) << 2;
            float4 v = *(const float4*)(hbase + (size_t)row * HIDDEN + kc + c4);
            *(float4*)(&As[0][0] + row * AS_STRIDE + c4) = v;
        }
        __syncthreads();
#pragma unroll 8
        for (int kk = 0; kk < KCHUNK; kk += 4) {
            v2f b  = *(const v2f*)(wrow + kc + kk);
            v2f a0 = *(const v2f*)(&As[0][0] + (     lm) * AS_STRIDE + kk + 2 * lh);
            v2f a1 = *(const v2f*)(&As[0][0] + (16 + lm) * AS_STRIDE + kk + 2 * lh);
            acc0 = __builtin_amdgcn_wmma_f32_16x16x4_f32(false, a0, false, b,
                                                         (short)0, acc0, false, false);
            acc1 = __builtin_amdgcn_wmma_f32_16x16x4_f32(false, a1, false, b,
                                                         (short)0, acc1, false, false);
        }
    }
#endif

    __syncthreads();
    // C layout: VGPR r, lanes 0-15 -> (M=r, N=lane); lanes 16-31 -> (M=8+r, N=lane-16)
#pragma unroll
    for (int r = 0; r < 8; ++r) {
        int tok0 = lh * 8 + r;
        Sc[tok0        * SC_STRIDE + n0 + lm] = 1.f / (1.f + __expf(-acc0[r]));
        Sc[(16 + tok0) * SC_STRIDE + n0 + lm] = 1.f / (1.f + __expf(-acc1[r]));
    }
    __syncthreads();

    // Top-8 selection: each wave handles 4 tokens; lane owns experts
    // {lane, lane+32, lane+64, lane+96} of the bias-corrected scores.
    for (int tl = 0; tl < 4; ++tl) {
        const int tok = wave * 4 + tl;
        float s0 = Sc[tok * SC_STRIDE + lane     ] + Bs[lane     ];
        float s1 = Sc[tok * SC_STRIDE + lane + 32] + Bs[lane + 32];
        float s2 = Sc[tok * SC_STRIDE + lane + 64] + Bs[lane + 64];
        float s3 = Sc[tok * SC_STRIDE + lane + 96] + Bs[lane + 96];
        float sum = 0.f;
#pragma unroll
        for (int i = 0; i < TOPK; ++i) {
            // local argmax (ascending expert index => stable tie-break)
            float bv = s0; int bi = lane;
            if (s1 > bv) { bv = s1; bi = lane + 32; }
            if (s2 > bv) { bv = s2; bi = lane + 64; }
            if (s3 > bv) { bv = s3; bi = lane + 96; }
            // wave-wide butterfly argmax, lower index wins ties (jax top_k)
#pragma unroll
            for (int off = 16; off > 0; off >>= 1) {
                float ov = __shfl_xor(bv, off, 32);
                int   oi = __shfl_xor(bi, off, 32);
                if (ov > bv || (ov == bv && oi < bi)) { bv = ov; bi = oi; }
            }
            float rw = Sc[tok * SC_STRIDE + bi];   // raw sigmoid weight
            sum += rw;
            if (lane == 0) { PkI[tok][i] = bi; PkW[tok][i] = rw; }
            if ((bi & 31) == lane) {               // knock out the winner
                int j = bi >> 5;
                s0 = (j == 0) ? -3.402823466e38f : s0;
                s1 = (j == 1) ? -3.402823466e38f : s1;
                s2 = (j == 2) ? -3.402823466e38f : s2;
                s3 = (j == 3) ? -3.402823466e38f : s3;
            }
        }
        float inv = 1.f / (sum + 1e-20f);          // NORM_TOPK_PROB, scaling=1.0
        if (lane < TOPK) {
            int g = m0 + tok;
            out_idx[(size_t)g * TOPK + lane] = PkI[tok][lane];
            out_w  [(size_t)g * TOPK + lane] = PkW[tok][lane] * inv;
        }
    }
}

extern "C" void kernel_launch(void* const* d_in, const int* in_sizes, int n_in,
                              void* d_out, int out_size, void* d_ws, size_t ws_size,
                              hipStream_t stream) {
    const float* hs   = (const float*)d_in[0];   // [4,4096,4096] fp32
    const float* wgt  = (const float*)d_in[1];   // [128,4096]    fp32
    const float* bias = (const float*)d_in[2];   // [128]         fp32

    const int T = in_sizes[0] / HIDDEN;          // 16384 tokens

    // d_out = [int32 topk_indices (T*8)] ++ [float topk_weights (T*8)]
    int*   out_idx = (int*)d_out;
    float* out_w   = (float*)d_out + (size_t)T * TOPK;

    dim3 grid(T / BM), block(256);
    hipLaunchKernelGGL(glm4v_moe_router_kernel, grid, block, 0, stream,
                       hs, wgt, bias, out_idx, out_w);
}